// QLSTM_65481071404374
// MI455X (gfx1250) — compile-verified
//
#include <hip/hip_runtime.h>
#include <hip/hip_bf16.h>

typedef float v2f __attribute__((ext_vector_type(2)));
typedef float v8f __attribute__((ext_vector_type(8)));

#define TT 2048
#define BB 256
#define DD 128
#define HQ 4
#define MROWS (TT * BB)        // 524288 GEMM rows
#define NCOLS 16               // 4 gates x 4 qubits

// GEMM tile geometry
#define BLK_ROWS  64           // X rows per workgroup (4 waves x 16)
#define THREADS   128
#define LDS_STRIDE 132         // padded float stride -> conflict-free ds_read_b64

// =====================================================================
// Phase 1: PRE[T*B, 16] = X[T*B,128] @ Wcat[128,16] + (b + theta)
// using V_WMMA_F32_16X16X4_F32 (one 16x16 N-tile, K walked in steps of 4)
// =====================================================================
__global__ __launch_bounds__(THREADS)
void qlstm_gemm_wmma(const float* __restrict__ X,
                     const float* __restrict__ Wf, const float* __restrict__ bf, const float* __restrict__ thf,
                     const float* __restrict__ Wi, const float* __restrict__ bi, const float* __restrict__ thi,
                     const float* __restrict__ Wu, const float* __restrict__ bu, const float* __restrict__ thu,
                     const float* __restrict__ Wo, const float* __restrict__ bo, const float* __restrict__ tho,
                     float* __restrict__ pre)
{
    __shared__ float sX[BLK_ROWS * LDS_STRIDE];

    const int tid  = threadIdx.x;
    const int wave = tid >> 5;
    const int lane = tid & 31;
    const int lo   = lane & 15;   // M row (A) / N col (B,C,D)
    const int kh   = lane >> 4;   // K half-select per ISA f32 layouts

    const size_t rowBase = (size_t)blockIdx.x * BLK_ROWS;

    // ---- stage 64x128 f32 X tile into LDS: fully coalesced float4 loads ----
    {
        const float4* g4 = (const float4*)(X + rowBase * DD);
        #pragma unroll
        for (int it = 0; it < (BLK_ROWS * DD / 4) / THREADS; ++it) {
            int idx = it * THREADS + tid;     // float4 index within tile
            int r   = idx >> 5;               // 32 float4 per 128-float row
            int c4  = idx & 31;
            float4 v = g4[idx];
            *((float4*)&sX[r * LDS_STRIDE + c4 * 4]) = v;
        }
    }

    // ---- per-lane B fragments + fused bias (weights are 8KB, cache-hot) ----
    // column n = gate*4 + j ; B[k][n] = W_gate[k*4 + j]
    const int gate = lo >> 2;
    const int j    = lo & 3;
    const float* Wg = (gate == 0) ? Wf : (gate == 1) ? Wi : (gate == 2) ? Wu : Wo;
    const float* bg = (gate == 0) ? bf : (gate == 1) ? bi : (gate == 2) ? bu : bo;
    const float* tg = (gate == 0) ? thf: (gate == 1) ? thi: (gate == 2) ? thu: tho;

    v2f bfr[32];
    #pragma unroll
    for (int kb = 0; kb < 32; ++kb) {
        int k0 = kb * 4 + kh * 2;                 // B: v0 holds K=k0, v1 holds K=k0+1
        bfr[kb][0] = Wg[(k0 + 0) * HQ + j];
        bfr[kb][1] = Wg[(k0 + 1) * HQ + j];
    }
    const float bias = bg[j] + tg[j];             // fold b + theta into accumulator

    v8f acc;
    #pragma unroll
    for (int i = 0; i < 8; ++i) acc[i] = bias;    // bias uniform down each column

    __syncthreads();

    // ---- K loop: 32 chained V_WMMA_F32_16X16X4_F32 ----
    // A layout (16x4 f32): lane lo -> row M=lo ; v0 = K=kh*2, v1 = K=kh*2+1
    const float* arow = &sX[(wave * 16 + lo) * LDS_STRIDE + kh * 2];
    #pragma unroll
    for (int kb = 0; kb < 32; ++kb) {
        v2f a = *(const v2f*)(arow + kb * 4);     // 8B-aligned ds_read_b64
        acc = __builtin_amdgcn_wmma_f32_16x16x4_f32(
                  /*neg_a=*/false, a, /*neg_b=*/false, bfr[kb],
                  /*c_mod=*/(short)0, acc, /*reuse_a=*/false, /*reuse_b=*/false);
    }

    // ---- store D per C/D layout: VGPR i -> M = i + 8*kh, N = lo ----
    #pragma unroll
    for (int i = 0; i < 8; ++i) {
        size_t row = rowBase + (size_t)wave * 16 + kh * 8 + i;
        pre[row * NCOLS + lo] = acc[i];
    }
}

// =====================================================================
// Phase 2: sequential LSTM recurrence, one lane per batch element.
// PRE stream (33.5MB) is L2-resident; hidden weights live in VGPRs.
// =====================================================================
__device__ __forceinline__ float fast_sigmoid(float x) {
    return __builtin_amdgcn_rcpf(1.0f + __expf(-x));
}
__device__ __forceinline__ float fast_tanh(float x) {
    // 1 - 2/(e^{2x}+1): saturates cleanly to +-1 at both extremes
    return 1.0f - 2.0f * __builtin_amdgcn_rcpf(__expf(2.0f * x) + 1.0f);
}

__global__ void qlstm_recurrence(const float* __restrict__ pre,
                                 const float* __restrict__ Wf, const float* __restrict__ Wi,
                                 const float* __restrict__ Wu, const float* __restrict__ Wo,
                                 float* __restrict__ out)
{
    const int b = blockIdx.x * blockDim.x + threadIdx.x;
    if (b >= BB) return;

    // hidden-recurrence weights: rows 128..131 of each W, kept in registers
    const float* Ws[4] = {Wf, Wi, Wu, Wo};
    float wh[4][4][4];
    #pragma unroll
    for (int g = 0; g < 4; ++g)
        #pragma unroll
        for (int k = 0; k < 4; ++k)
            #pragma unroll
            for (int q = 0; q < 4; ++q)
                wh[g][k][q] = Ws[g][(DD + k) * HQ + q];

    float h[4] = {0.f, 0.f, 0.f, 0.f};
    float c[4] = {0.f, 0.f, 0.f, 0.f};

    float* hseq = out;                                   // [T,B,4]
    float* hx   = out + (size_t)TT * BB * HQ;            // [B,4]
    float* cx   = hx + (size_t)BB * HQ;                  // [B,4]

    const float4* p4 = (const float4*)pre;               // 4 float4 per (t,b)

    size_t i0 = (size_t)b * 4;
    float4 cur0 = p4[i0 + 0], cur1 = p4[i0 + 1], cur2 = p4[i0 + 2], cur3 = p4[i0 + 3];

    for (int t = 0; t < TT; ++t) {
        // software prefetch of next step's pre-activations (clamped at end)
        int tn = (t + 1 < TT) ? (t + 1) : t;
        size_t ni = ((size_t)tn * BB + b) * 4;
        float4 nxt0 = p4[ni + 0], nxt1 = p4[ni + 1], nxt2 = p4[ni + 2], nxt3 = p4[ni + 3];

        float pr[16] = {cur0.x, cur0.y, cur0.z, cur0.w,
                        cur1.x, cur1.y, cur1.z, cur1.w,
                        cur2.x, cur2.y, cur2.z, cur2.w,
                        cur3.x, cur3.y, cur3.z, cur3.w};

        // qlayer per gate: cumprod_j cos(pre + h @ Wh)   (b+theta already fused)
        float gv[4][4];
        #pragma unroll
        for (int g = 0; g < 4; ++g) {
            float cp = 1.0f;
            #pragma unroll
            for (int q = 0; q < 4; ++q) {
                float z = pr[g * 4 + q];
                #pragma unroll
                for (int k = 0; k < 4; ++k)
                    z = __fmaf_rn(h[k], wh[g][k][q], z);
                cp *= __cosf(z);
                gv[g][q] = cp;
            }
        }

        #pragma unroll
        for (int q = 0; q < 4; ++q) {
            float fg = fast_sigmoid(gv[0][q]);
            float ig = fast_sigmoid(gv[1][q]);
            float gg = fast_tanh(gv[2][q]);
            float og = fast_sigmoid(gv[3][q]);
            c[q] = fg * c[q] + ig * gg;
            h[q] = og * fast_tanh(c[q]);
        }

        ((float4*)hseq)[(size_t)t * BB + b] = make_float4(h[0], h[1], h[2], h[3]);

        cur0 = nxt0; cur1 = nxt1; cur2 = nxt2; cur3 = nxt3;
    }

    ((float4*)hx)[b] = make_float4(h[0], h[1], h[2], h[3]);
    ((float4*)cx)[b] = make_float4(c[0], c[1], c[2], c[3]);
}

// =====================================================================
extern "C" void kernel_launch(void* const* d_in, const int* in_sizes, int n_in,
                              void* d_out, int out_size, void* d_ws, size_t ws_size,
                              hipStream_t stream)
{
    const float* X   = (const float*)d_in[0];
    const float* Wf  = (const float*)d_in[1];
    const float* bf  = (const float*)d_in[2];
    const float* thf = (const float*)d_in[3];
    const float* Wi  = (const float*)d_in[4];
    const float* bi  = (const float*)d_in[5];
    const float* thi = (const float*)d_in[6];
    const float* Wu  = (const float*)d_in[7];
    const float* bu  = (const float*)d_in[8];
    const float* thu = (const float*)d_in[9];
    const float* Wo  = (const float*)d_in[10];
    const float* bo  = (const float*)d_in[11];
    const float* tho = (const float*)d_in[12];

    float* pre = (float*)d_ws;   // [T*B, 16] f32 = 33.5 MB scratch

    qlstm_gemm_wmma<<<dim3(MROWS / BLK_ROWS), dim3(THREADS), 0, stream>>>(
        X, Wf, bf, thf, Wi, bi, thi, Wu, bu, thu, Wo, bo, tho, pre);

    qlstm_recurrence<<<dim3(BB / 32), dim3(32), 0, stream>>>(
        pre, Wf, Wi, Wu, Wo, (float*)d_out);
}